// LarsField_33200097199040
// MI455X (gfx1250) — compile-verified
//
#include <hip/hip_runtime.h>
#include <hip/hip_bf16.h>
#include <math.h>

typedef __attribute__((ext_vector_type(2))) float v2f;
typedef __attribute__((ext_vector_type(8))) float v8f;

// Exact 32-lane f32 sum using the matrix engine (V_WMMA_F32_16X16X4_F32).
// A: per-lane partial in VGPR0, 0 in VGPR1 -> A[m][k] has x_m at k=0 (lanes 0-15)
// and x_{m+16} at k=2 (lanes 16-31), zeros elsewhere. B = all-ones (layout-proof).
// D[m][n] = x_m + x_{m+16} for every n; each lane's 8 D regs cover 8 distinct rows
// of one half-wave, so sum-of-8 + shfl_xor(16) gives the full wave total in f32.
__device__ __forceinline__ float wave_sum_wmma(float x) {
    v2f a; a.x = x;    a.y = 0.0f;
    v2f b; b.x = 1.0f; b.y = 1.0f;
    v8f c = {};
    c = __builtin_amdgcn_wmma_f32_16x16x4_f32(false, a, false, b, (short)0, c,
                                              false, false);
    float s = c[0] + c[1] + c[2] + c[3] + c[4] + c[5] + c[6] + c[7];
    s += __shfl_xor(s, 16, 32);
    return s;
}

__global__ void lars_init_ws(float* __restrict__ ws) {
    if (threadIdx.x < 4) ws[threadIdx.x] = 0.0f;
}

// Streaming reduction: S = sum w_i, WX = sum w_i * X_i  (w_i = exp(-50*||r-X_i||^2))
__global__ void __launch_bounds__(256)
lars_partial(const float* __restrict__ X, const float* __restrict__ r,
             float* __restrict__ ws, int n) {
    const float rx = r[0], ry = r[1], rz = r[2];

    float S = 0.0f, wxs = 0.0f, wys = 0.0f, wzs = 0.0f;

    auto acc = [&](float x, float y, float z) {
        float dx = x - rx, dy = y - ry, dz = z - rz;
        float sq = fmaf(dx, dx, fmaf(dy, dy, dz * dz));
        float w  = __expf(-50.0f * sq);          // 1/(2h^2) = 50, h = 0.1
        S   += w;
        wxs  = fmaf(w, x, wxs);
        wys  = fmaf(w, y, wys);
        wzs  = fmaf(w, z, wzs);
    };

    const long tid    = (long)blockIdx.x * blockDim.x + threadIdx.x;
    const long stride = (long)gridDim.x * blockDim.x;
    const long nquad  = (long)n >> 2;            // 4 samples = 48B = 3 x float4
    const float4* X4  = (const float4*)X;

    for (long q = tid; q < nquad; q += stride) {
        __builtin_prefetch((const void*)(X4 + 3 * (q + stride)), 0, 1);
        float4 p0 = X4[3 * q + 0];
        float4 p1 = X4[3 * q + 1];
        float4 p2 = X4[3 * q + 2];
        acc(p0.x, p0.y, p0.z);
        acc(p0.w, p1.x, p1.y);
        acc(p1.z, p1.w, p2.x);
        acc(p2.y, p2.z, p2.w);
    }
    // tail (n not divisible by 4)
    for (long i = (nquad << 2) + tid; i < (long)n; i += stride) {
        acc(X[3 * i + 0], X[3 * i + 1], X[3 * i + 2]);
    }

    // Wave reduction on the matrix engine (EXEC is all-ones here: no early exits).
    float tS = wave_sum_wmma(S);
    float tx = wave_sum_wmma(wxs);
    float ty = wave_sum_wmma(wys);
    float tz = wave_sum_wmma(wzs);

    __shared__ float sacc[8][4];                 // 256 threads -> 8 wave32s
    const int lane = threadIdx.x & 31;
    const int wid  = threadIdx.x >> 5;
    if (lane == 0) {
        sacc[wid][0] = tS; sacc[wid][1] = tx;
        sacc[wid][2] = ty; sacc[wid][3] = tz;
    }
    __syncthreads();
    if (threadIdx.x < 4) {
        float v = 0.0f;
        const int nw = blockDim.x >> 5;
        for (int w = 0; w < nw; ++w) v += sacc[w][threadIdx.x];
        atomicAdd(&ws[threadIdx.x], v);
    }
}

__global__ void lars_finalize(const float* __restrict__ ws,
                              const float* __restrict__ r,
                              const float* __restrict__ g,
                              const float* __restrict__ fdg,
                              float* __restrict__ out, int n) {
    int c = threadIdx.x;
    if (c >= 3) return;
    const float  S    = ws[0];
    const float  WX   = ws[1 + c];
    const double h2   = 0.01;                    // h = 0.1
    // log_norm = -log(N) - (d/2) log(2*pi*h^2), d = 3
    const float norm  = (float)exp(-log((double)n) - 1.5 * log(2.0 * M_PI * h2));
    const float fhat  = norm * S;
    // d f_hat / dr_c = -(norm/h^2) * (S*r_c - WX_c)
    const float grad  = -(norm / (float)h2) * (S * r[c] - WX);
    const float phi   = grad - fdg[c];
    const float V     = -1.0f * phi;             // D_COEF = 1
    out[c] = (V + g[c]) / fhat;
}

extern "C" void kernel_launch(void* const* d_in, const int* in_sizes, int n_in,
                              void* d_out, int out_size, void* d_ws, size_t ws_size,
                              hipStream_t stream) {
    const float* r   = (const float*)d_in[0];
    const float* X   = (const float*)d_in[1];
    const float* g   = (const float*)d_in[2];
    const float* fdg = (const float*)d_in[3];
    float* out       = (float*)d_out;
    float* ws        = (float*)d_ws;

    const int n = in_sizes[1] / 3;               // X is (N, 3) f32

    lars_init_ws<<<1, 32, 0, stream>>>(ws);

    const int  threads = 256;
    long need = ((long)n / 4 + threads - 1) / threads;
    int blocks = (int)(need < 1 ? 1 : (need > 2048 ? 2048 : need));
    lars_partial<<<blocks, threads, 0, stream>>>(X, r, ws, n);

    lars_finalize<<<1, 4, 0, stream>>>(ws, r, g, fdg, out, n);
}